// GCNConv_59390807769606
// MI455X (gfx1250) — compile-verified
//
#include <hip/hip_runtime.h>

#define D_FEAT 128

// ---------------------------------------------------------------------------
// Zeroing kernels (harness poisons d_out / d_ws with 0xAA before timing).
// ---------------------------------------------------------------------------
__global__ void zero_f32_kernel(float* __restrict__ p, int n) {
    int i = blockIdx.x * blockDim.x + threadIdx.x;
    if (i < n) p[i] = 0.0f;
}

__global__ void zero_u32_kernel(unsigned* __restrict__ p, int n) {
    int i = blockIdx.x * blockDim.x + threadIdx.x;
    if (i < n) p[i] = 0u;
}

// ---------------------------------------------------------------------------
// Pass 1: out-degree histogram over src (native global_atomic_add_u32, L2-hot)
// ---------------------------------------------------------------------------
__global__ void deg_kernel(const int* __restrict__ src,
                           unsigned* __restrict__ deg,
                           int n_edges) {
    int i = blockIdx.x * blockDim.x + threadIdx.x;
    if (i < n_edges) {
        atomicAdd(&deg[src[i]], 1u);
    }
}

// ---------------------------------------------------------------------------
// Pass 2: rs[n] = rsqrt(max(deg, 1))   (v_rsq_f32)
// ---------------------------------------------------------------------------
__global__ void rs_kernel(const unsigned* __restrict__ deg,
                          float* __restrict__ rs,
                          int n_nodes) {
    int i = blockIdx.x * blockDim.x + threadIdx.x;
    if (i < n_nodes) {
        float d = (float)deg[i];
        rs[i] = __frsqrt_rn(fmaxf(d, 1.0f));
    }
}

// ---------------------------------------------------------------------------
// Pass 3: gather - scale - scatter-add, 32 edges per wave32.
//
// Per wave:
//   * 2x coalesced global_load_b32: endpoints of 32 edges (lane = edge).
//   * 2x per-lane gathers from rs[] -> all 32 edge norms computed vectorized
//     up front (normv); no memory access for norms inside the edge loop.
//   * Edge loop: v_readlane_b32 scalarizes {s, d, norm} for edge i, then
//       - 512B row gather: SADDR-form global_load_b128, voffset = lane*16,
//       - 4x no-return global_atomic_add_f32 (RMW resolved in the 192MB L2;
//         out = 51MB, x = 51MB -> fully L2-resident working set).
//   * Full blocks (the only case for E % 32 == 0) take an unroll-4 path so
//     the scheduler can clause independent row loads and overlap their L2
//     latency with preceding iterations' atomics.
// ---------------------------------------------------------------------------
__global__ __launch_bounds__(256) void scatter_kernel(
    const float* __restrict__ x,
    const int*   __restrict__ src,
    const int*   __restrict__ dst,
    const float* __restrict__ rs,
    float*       __restrict__ out,
    int n_edges) {

    const int lane = threadIdx.x & 31;
    const int wave = blockIdx.x * (256 >> 5) + (threadIdx.x >> 5);
    const int base = wave << 5;                    // 32 edges per wave
    if (base >= n_edges) return;

    // Coalesced endpoint loads for this wave's 32 edges (clamped tail).
    const int eidx = min(base + lane, n_edges - 1);
    const int vsrc = src[eidx];
    const int vdst = dst[eidx];

    // All 32 edge norms, computed vectorized (lane = edge).
    const float normv = rs[vsrc] * rs[vdst];

    const int cnt     = min(32, n_edges - base);
    const int byteoff = lane << 4;                 // lane*16 bytes into the row

#define GCN_EDGE_BODY(i)                                                            \
    {                                                                               \
        const int   s    = __builtin_amdgcn_readlane(vsrc, (i));                    \
        const int   d    = __builtin_amdgcn_readlane(vdst, (i));                    \
        const float norm = __uint_as_float(                                         \
            __builtin_amdgcn_readlane(__float_as_uint(normv), (i)));               \
        const float4 v = *(const float4*)((const char*)x +                          \
                                          (size_t)s * (D_FEAT * 4) + byteoff);      \
        float* o = (float*)((char*)out + (size_t)d * (D_FEAT * 4) + byteoff);       \
        __hip_atomic_fetch_add(o + 0, v.x * norm, __ATOMIC_RELAXED,                 \
                               __HIP_MEMORY_SCOPE_AGENT);                           \
        __hip_atomic_fetch_add(o + 1, v.y * norm, __ATOMIC_RELAXED,                 \
                               __HIP_MEMORY_SCOPE_AGENT);                           \
        __hip_atomic_fetch_add(o + 2, v.z * norm, __ATOMIC_RELAXED,                 \
                               __HIP_MEMORY_SCOPE_AGENT);                           \
        __hip_atomic_fetch_add(o + 3, v.w * norm, __ATOMIC_RELAXED,                 \
                               __HIP_MEMORY_SCOPE_AGENT);                           \
    }

    if (cnt == 32) {
        // Full block: unroll so independent row loads can be claused and
        // their L2 latency overlapped with earlier iterations' atomics.
#pragma unroll 4
        for (int i = 0; i < 32; ++i) {
            GCN_EDGE_BODY(i)
        }
    } else {
        for (int i = 0; i < cnt; ++i) {
            GCN_EDGE_BODY(i)
        }
    }
#undef GCN_EDGE_BODY
}

// ---------------------------------------------------------------------------
// Host-side launcher
// ---------------------------------------------------------------------------
extern "C" void kernel_launch(void* const* d_in, const int* in_sizes, int n_in,
                              void* d_out, int out_size, void* d_ws, size_t ws_size,
                              hipStream_t stream) {
    const float* x   = (const float*)d_in[0];
    const int*   src = (const int*)d_in[1];
    const int*   dst = (const int*)d_in[2];
    float*       out = (float*)d_out;

    const int n_nodes = in_sizes[0] / D_FEAT;
    const int n_edges = in_sizes[1];

    // Workspace layout: [ deg: u32 x n_nodes | rs: f32 x n_nodes ]
    unsigned* deg = (unsigned*)d_ws;
    float*    rs  = (float*)((char*)d_ws + (size_t)n_nodes * sizeof(unsigned));

    // Zero accumulators (d_out / d_ws are poisoned by the harness).
    zero_f32_kernel<<<(out_size + 255) / 256, 256, 0, stream>>>(out, out_size);
    zero_u32_kernel<<<(n_nodes + 255) / 256, 256, 0, stream>>>(deg, n_nodes);

    // Pass 1: degree histogram.
    deg_kernel<<<(n_edges + 255) / 256, 256, 0, stream>>>(src, deg, n_edges);

    // Pass 2: per-node rsqrt(max(deg,1)).
    rs_kernel<<<(n_nodes + 255) / 256, 256, 0, stream>>>(deg, rs, n_nodes);

    // Pass 3: 32 edges per wave, 8 waves per 256-thread block.
    const int waves  = (n_edges + 31) / 32;
    const int blocks = (waves + 7) / 8;
    scatter_kernel<<<blocks, 256, 0, stream>>>(x, src, dst, rs, out, n_edges);
}